// LLAConv2d_51393578664092
// MI455X (gfx1250) — compile-verified
//
#include <hip/hip_runtime.h>
#include <stdint.h>

// LLAConv2d: per-sample mixed-kernel 1x1 conv == batch of 64x25600x64 fp32 GEMMs.
// fp32 WMMA (v_wmma_f32_16x16x4_f32); weights staged to LDS via CDNA5 async
// global->LDS copy; x streamed from HBM exactly once (bandwidth-bound, AI~16).

typedef __attribute__((ext_vector_type(2))) float v2f;
typedef __attribute__((ext_vector_type(4))) float v4f;
typedef __attribute__((ext_vector_type(8))) float v8f;

#define NB   32
#define NE   8
#define CIN  64
#define COUT 64
#define HW   25600
#define KSZ  (COUT * CIN)   /* 4096 floats per batch */
#define SKS  68             /* padded LDS row stride (dwords): 68 % 64 == 4 -> conflict-free A reads */

// ---------------------------------------------------------------------------
// Kernel 1: K[b] = sum_e alpha[b,e] * KE[e]   (32 x 64 x 64 fp32 -> workspace)
// ---------------------------------------------------------------------------
__global__ __launch_bounds__(256) void lla_mix_kernels(
    const float* __restrict__ alpha,   // [NB, NE]
    const float* __restrict__ ke,      // [NE, COUT, CIN]
    float* __restrict__ kmix)          // [NB, COUT, CIN]
{
    const int b = blockIdx.x;
    __shared__ float a[NE];
    if (threadIdx.x < NE) a[threadIdx.x] = alpha[b * NE + threadIdx.x];
    __syncthreads();

    for (int v = threadIdx.x; v < KSZ / 4; v += 256) {
        v4f s = {0.f, 0.f, 0.f, 0.f};
#pragma unroll
        for (int e = 0; e < NE; ++e) {
            v4f kv = *(const v4f*)(ke + e * KSZ + v * 4);
            s += a[e] * kv;
        }
        *(v4f*)(kmix + b * KSZ + v * 4) = s;
    }
}

// ---------------------------------------------------------------------------
// Kernel 2: out[b,o,p] = sum_i K[b,o,i] * x[b,i,p]
// Block = (256 pixels, one batch); 8 waves x 32 pixels (2 N-tiles); each wave
// does all 64 Cout rows as 4 WMMA M-tiles; A-fragments shared by both N-tiles.
//
// WMMA fp32 fragment layout (ISA 7.12.2):
//   A 16x4 : VGPR0 = A[M=lane%16][K=2*half], VGPR1 = A[..][K=2*half+1]
//   B 4x16 : VGPR0 = B[K=2*half][N=lane%16], VGPR1 = B[K=2*half+1][..]
//   D 16x16: VGPRj = D[M = 8*half + j][N = lane%16]
// ---------------------------------------------------------------------------
__global__ __launch_bounds__(256) void lla_conv_wmma(
    const float* __restrict__ x,       // [NB, CIN, HW]
    const float* __restrict__ kmix,    // [NB, COUT, CIN]
    float* __restrict__ out)           // [NB, COUT, HW]
{
    __shared__ float sK[COUT * SKS];

    const int b   = blockIdx.y;
    const int tid = threadIdx.x;

    // ---- Stage K[b] into LDS (padded rows) with CDNA5 async global->LDS ----
    // GVS mode: saddr = kmix[b] base, vaddr = 32-bit byte offset. The LDS dest
    // operand is the low 32 bits of the element's flat address: this is the
    // correct DS-style LDS address AND a ptrtoint escape of sK, so LLVM must
    // treat the memory-clobbering asm as a writer of sK (keeps the ds_load
    // A-fragment reads alive -- round-2 regression fix).
    {
        const unsigned long long kb = (unsigned long long)(kmix + b * KSZ);
#pragma unroll
        for (int s = 0; s < 4; ++s) {
            int v   = tid + 256 * s;        // float4 index, 0..1023
            int row = v >> 4;               // 16 float4 per 64-float row
            int c4  = (v & 15) << 2;
            unsigned int lds_addr = (unsigned int)(uintptr_t)&sK[row * SKS + c4];
            unsigned int g_off    = (unsigned int)(v << 4);
            asm volatile("global_load_async_to_lds_b128 %0, %1, %2"
                         :
                         : "v"(lds_addr), "v"(g_off), "s"(kb)
                         : "memory");
        }
        asm volatile("s_wait_asynccnt 0x0" ::: "memory");
    }
    __syncthreads();

    const int wave = tid >> 5;
    const int lane = tid & 31;
    const int l16  = lane & 15;
    const int half = lane >> 4;
    const int p0   = blockIdx.x * 256 + wave * 32 + l16;  // N-tile 0 pixel
    const int p1   = p0 + 16;                             // N-tile 1 pixel

    const float* xb = x + (size_t)b * CIN * HW;

    v8f acc[8] = {};   // [tile][m]: acc[m] = tile0, acc[4+m] = tile1

#pragma unroll
    for (int k0 = 0; k0 < CIN; k0 += 4) {
        const int kk = k0 + 2 * half;

        // B fragments: x dwords, half-wave-contiguous in pixels (64B segments)
        v2f B0, B1;
        B0.x = xb[(size_t)kk * HW + p0];
        B0.y = xb[(size_t)(kk + 1) * HW + p0];
        B1.x = xb[(size_t)kk * HW + p1];
        B1.y = xb[(size_t)(kk + 1) * HW + p1];

        // A fragments: consecutive K pair -> ds_load_b64, bank-conflict-free;
        // shared by both N-tiles (halves LDS reads per wmma).
        v2f A0 = *(const v2f*)(&sK[(0 * 16 + l16) * SKS + kk]);
        v2f A1 = *(const v2f*)(&sK[(1 * 16 + l16) * SKS + kk]);
        v2f A2 = *(const v2f*)(&sK[(2 * 16 + l16) * SKS + kk]);
        v2f A3 = *(const v2f*)(&sK[(3 * 16 + l16) * SKS + kk]);

        acc[0] = __builtin_amdgcn_wmma_f32_16x16x4_f32(false, A0, false, B0, (short)0, acc[0], false, false);
        acc[1] = __builtin_amdgcn_wmma_f32_16x16x4_f32(false, A1, false, B0, (short)0, acc[1], false, false);
        acc[2] = __builtin_amdgcn_wmma_f32_16x16x4_f32(false, A2, false, B0, (short)0, acc[2], false, false);
        acc[3] = __builtin_amdgcn_wmma_f32_16x16x4_f32(false, A3, false, B0, (short)0, acc[3], false, false);
        acc[4] = __builtin_amdgcn_wmma_f32_16x16x4_f32(false, A0, false, B1, (short)0, acc[4], false, false);
        acc[5] = __builtin_amdgcn_wmma_f32_16x16x4_f32(false, A1, false, B1, (short)0, acc[5], false, false);
        acc[6] = __builtin_amdgcn_wmma_f32_16x16x4_f32(false, A2, false, B1, (short)0, acc[6], false, false);
        acc[7] = __builtin_amdgcn_wmma_f32_16x16x4_f32(false, A3, false, B1, (short)0, acc[7], false, false);
    }

    // Scatter D tiles: row m*16 + 8*half + j, columns p0 / p1.
    float* ob0 = out + (size_t)b * COUT * HW + p0;
    float* ob1 = out + (size_t)b * COUT * HW + p1;
#pragma unroll
    for (int m = 0; m < 4; ++m) {
#pragma unroll
        for (int j = 0; j < 8; ++j) {
            ob0[(size_t)(m * 16 + 8 * half + j) * HW] = acc[m][j];
            ob1[(size_t)(m * 16 + 8 * half + j) * HW] = acc[4 + m][j];
        }
    }
}

// ---------------------------------------------------------------------------
extern "C" void kernel_launch(void* const* d_in, const int* in_sizes, int n_in,
                              void* d_out, int out_size, void* d_ws, size_t ws_size,
                              hipStream_t stream) {
    const float* x     = (const float*)d_in[0];   // [32,64,160,160]
    const float* alpha = (const float*)d_in[1];   // [32,8]
    const float* ke    = (const float*)d_in[2];   // [8,64,64,1,1]
    float*       out   = (float*)d_out;           // [32,64,160,160]
    float*       kmix  = (float*)d_ws;            // [32,64,64] = 512 KB scratch

    lla_mix_kernels<<<NB, 256, 0, stream>>>(alpha, ke, kmix);
    lla_conv_wmma<<<dim3(HW / 256, NB), 256, 0, stream>>>(x, kmix, out);
}